// GAT_68762426409265
// MI455X (gfx1250) — compile-verified
//
#include <hip/hip_runtime.h>
#include <hip/hip_bf16.h>

// ---------------------------------------------------------------------------
// 2-layer GAT for MI455X (gfx1250, wave32).
//   L1: h1 = x @ W1 (fp32 WMMA 16x16x4), s/d scores, segment softmax over dst,
//       edge-weighted scatter-add -> out1 ; ReLU fused into L2 GEMM A-load.
//   L2: h2 = relu(out1) @ W2, same edge pipeline -> d_out [N,40].
// GEMMs are HBM-bound (x = 102MB); edge gathers hit 192MB L2 (h1=25.6MB).
// B is stored TRANSPOSED in LDS with a pitch giving bank = (4*col+k)%64 so
// each B operand is a single conflict-free ds_load_b64; each wave owns TWO
// 16-row tiles so every B load feeds two v_wmma_f32_16x16x4_f32.
// ---------------------------------------------------------------------------

typedef __attribute__((ext_vector_type(2))) float v2f;
typedef __attribute__((ext_vector_type(8))) float v8f;

#define NEG_SLOPE 0.2f

__device__ __forceinline__ unsigned f2ord(float f) {
    unsigned u = __float_as_uint(f);
    return (u & 0x80000000u) ? ~u : (u | 0x80000000u);
}
__device__ __forceinline__ float ord2f(unsigned k) {
    unsigned u = (k & 0x80000000u) ? (k & 0x7fffffffu) : ~k;
    return __uint_as_float(u);
}
__device__ __forceinline__ float leaky(float x) {
    return x > 0.0f ? x : NEG_SLOPE * x;
}

// ------------------------- GEMM 1: h1 = x @ W1  [N,K] x [K,64] --------------
// 8 waves/block; each wave computes TWO 16-row tiles x 4 col tiles.
// W staged transposed in LDS: sW[col*132 + k], 128 K-rows per phase (33.8KB).
#define P1 132
__global__ __launch_bounds__(256) void gemm1_kernel(
    const float* __restrict__ X, const float* __restrict__ W,
    float* __restrict__ H, int N, int K) {
    __shared__ float sW[64 * P1];
    const int lane = threadIdx.x & 31;
    const int wave = threadIdx.x >> 5;
    const int half = lane >> 4;      // 0: lanes 0-15, 1: lanes 16-31
    const int col0 = lane & 15;
    const int nrt  = (N + 15) >> 4;
    const int rt0  = (blockIdx.x * 8 + wave) * 2;
    const int rt1  = rt0 + 1;
    const int rowA0 = (rt0 < nrt) ? min(rt0 * 16 + col0, N - 1) : 0;
    const int rowA1 = (rt1 < nrt) ? min(rt1 * 16 + col0, N - 1) : 0;

    v8f acc[2][4];
    #pragma unroll
    for (int u = 0; u < 2; ++u)
        #pragma unroll
        for (int t = 0; t < 4; ++t) acc[u][t] = v8f{};

    const int phases = K >> 7;  // K multiple of 128 (K=256 here)
    for (int ph = 0; ph < phases; ++ph) {
        // transpose-stage 128x64 chunk of W into LDS
        for (int i = threadIdx.x; i < 128 * 64; i += 256) {
            int r = i >> 6, c = i & 63;                 // r = K row, c = col
            sW[c * P1 + r] = W[(size_t)ph * 128 * 64 + i];
        }
        __syncthreads();

        const float* xr0 = X + (size_t)rowA0 * K + ph * 128;
        const float* xr1 = X + (size_t)rowA1 * K + ph * 128;
        for (int kk = 0; kk < 32; ++kk) {
            const int kb = kk * 4 + half * 2;           // local K pair
            v2f a0 = *(const v2f*)(xr0 + kb);
            v2f a1 = *(const v2f*)(xr1 + kb);
            #pragma unroll
            for (int t = 0; t < 4; ++t) {
                v2f b = *(const v2f*)(&sW[(t * 16 + col0) * P1 + kb]);
                acc[0][t] = __builtin_amdgcn_wmma_f32_16x16x4_f32(
                    false, a0, false, b, (short)0, acc[0][t], false, false);
                acc[1][t] = __builtin_amdgcn_wmma_f32_16x16x4_f32(
                    false, a1, false, b, (short)0, acc[1][t], false, false);
            }
        }
        __syncthreads();
    }

    #pragma unroll
    for (int u = 0; u < 2; ++u) {
        int rt = rt0 + u;
        if (rt >= nrt) continue;
        #pragma unroll
        for (int t = 0; t < 4; ++t)
            #pragma unroll
            for (int r = 0; r < 8; ++r) {
                int row = rt * 16 + r + half * 8;       // C/D: M=r (lo)/r+8 (hi)
                if (row < N) H[(size_t)row * 64 + t * 16 + col0] = acc[u][t][r];
            }
    }
}

// --------------- GEMM 2: h2 = relu(out1) @ W2  [N,64] x [64,40] -------------
#define P2 68
__global__ __launch_bounds__(256) void gemm2_kernel(
    const float* __restrict__ Hin, const float* __restrict__ W,
    float* __restrict__ Hout, int N, int NC) {
    __shared__ float sW[48 * P2];   // transposed, 40 cols zero-padded to 48
    const int lane = threadIdx.x & 31;
    const int wave = threadIdx.x >> 5;
    const int half = lane >> 4;
    const int col0 = lane & 15;
    const int nrt  = (N + 15) >> 4;
    const int rt0  = (blockIdx.x * 8 + wave) * 2;
    const int rt1  = rt0 + 1;
    const int rowA0 = (rt0 < nrt) ? min(rt0 * 16 + col0, N - 1) : 0;
    const int rowA1 = (rt1 < nrt) ? min(rt1 * 16 + col0, N - 1) : 0;

    for (int i = threadIdx.x; i < 64 * 48; i += 256) {
        int r = i / 48, c = i % 48;
        sW[c * P2 + r] = (c < NC) ? W[(size_t)r * NC + c] : 0.0f;
    }
    __syncthreads();

    v8f acc[2][3];
    #pragma unroll
    for (int u = 0; u < 2; ++u)
        #pragma unroll
        for (int t = 0; t < 3; ++t) acc[u][t] = v8f{};

    const float* xr0 = Hin + (size_t)rowA0 * 64;
    const float* xr1 = Hin + (size_t)rowA1 * 64;
    for (int kk = 0; kk < 16; ++kk) {
        const int kb = kk * 4 + half * 2;
        v2f a0 = *(const v2f*)(xr0 + kb);
        v2f a1 = *(const v2f*)(xr1 + kb);
        a0[0] = fmaxf(a0[0], 0.0f); a0[1] = fmaxf(a0[1], 0.0f);   // fused ReLU
        a1[0] = fmaxf(a1[0], 0.0f); a1[1] = fmaxf(a1[1], 0.0f);
        #pragma unroll
        for (int t = 0; t < 3; ++t) {
            v2f b = *(const v2f*)(&sW[(t * 16 + col0) * P2 + kb]);
            acc[0][t] = __builtin_amdgcn_wmma_f32_16x16x4_f32(
                false, a0, false, b, (short)0, acc[0][t], false, false);
            acc[1][t] = __builtin_amdgcn_wmma_f32_16x16x4_f32(
                false, a1, false, b, (short)0, acc[1][t], false, false);
        }
    }

    #pragma unroll
    for (int u = 0; u < 2; ++u) {
        int rt = rt0 + u;
        if (rt >= nrt) continue;
        #pragma unroll
        for (int t = 0; t < 3; ++t)
            #pragma unroll
            for (int r = 0; r < 8; ++r) {
                int row = rt * 16 + r + half * 8;
                int col = t * 16 + col0;
                if (row < N && col < NC)
                    Hout[(size_t)row * NC + col] = acc[u][t][r];
            }
    }
}

// ---------- per-node attention scores: s = h.a_src, d = h.a_dst -------------
__global__ __launch_bounds__(256) void sd_kernel(
    const float* __restrict__ H, const float* __restrict__ asrc,
    const float* __restrict__ adst, float* __restrict__ s,
    float* __restrict__ d, int N, int F) {
    const int wid  = (int)((blockIdx.x * (long long)blockDim.x + threadIdx.x) >> 5);
    const int lane = threadIdx.x & 31;
    if (wid >= N) return;
    float ps = 0.0f, pd = 0.0f;
    for (int f = lane; f < F; f += 32) {
        float hv = H[(size_t)wid * F + f];
        ps = fmaf(hv, asrc[f], ps);
        pd = fmaf(hv, adst[f], pd);
    }
    #pragma unroll
    for (int off = 16; off > 0; off >>= 1) {
        ps += __shfl_down(ps, off, 32);
        pd += __shfl_down(pd, off, 32);
    }
    if (lane == 0) { s[wid] = ps; d[wid] = pd; }
}

// ------------------------------ init helpers --------------------------------
__global__ void zero_f32(float* __restrict__ p, long long n) {
    long long i = blockIdx.x * (long long)blockDim.x + threadIdx.x;
    if (i < n) p[i] = 0.0f;
}
__global__ void init_seg(unsigned* __restrict__ m, float* __restrict__ denom, int n) {
    int i = blockIdx.x * blockDim.x + threadIdx.x;
    if (i < n) { m[i] = 0x007FFFFFu; /* f2ord(-inf) */ denom[i] = 0.0f; }
}

// ------------------------------- edge passes --------------------------------
__global__ __launch_bounds__(256) void edge_max_kernel(
    const int* __restrict__ ei, const float* __restrict__ s,
    const float* __restrict__ d, unsigned* __restrict__ m, int E, int EA) {
    int e = blockIdx.x * blockDim.x + threadIdx.x;
    if (e >= EA) return;
    int src, dst;
    if (e < E) { src = ei[e]; dst = ei[E + e]; } else { src = dst = e - E; }
    float sc = leaky(s[src] + d[dst]);
    atomicMax(&m[dst], f2ord(sc));
}

__global__ __launch_bounds__(256) void edge_expsum_kernel(
    const int* __restrict__ ei, const float* __restrict__ ew,
    const float* __restrict__ s, const float* __restrict__ d,
    const unsigned* __restrict__ m, float* __restrict__ denom,
    float* __restrict__ exw, int E, int EA) {
    int e = blockIdx.x * blockDim.x + threadIdx.x;
    if (e >= EA) return;
    int src, dst;
    float w;
    if (e < E) { src = ei[e]; dst = ei[E + e]; w = ew[e]; }
    else       { src = dst = e - E; w = 1.0f; }
    float sc = leaky(s[src] + d[dst]);
    float ex = __expf(sc - ord2f(m[dst]));
    exw[e] = ex * w;
    atomicAdd(&denom[dst], ex);
}

// one 64-thread group per edge; lanes cover features (F <= 64)
__global__ __launch_bounds__(256) void edge_agg_kernel(
    const int* __restrict__ ei, const float* __restrict__ exw,
    const float* __restrict__ denom, const float* __restrict__ H,
    float* __restrict__ out, int E, int EA, int F) {
    long long gid = blockIdx.x * (long long)blockDim.x + threadIdx.x;
    int e = (int)(gid >> 6);
    int f = (int)(gid & 63);
    if (e >= EA || f >= F) return;
    int src, dst;
    if (e < E) { src = ei[e]; dst = ei[E + e]; } else { src = dst = e - E; }
    float c = exw[e] / fmaxf(denom[dst], 1e-16f);
    atomicAdd(&out[(size_t)dst * F + f], H[(size_t)src * F + f] * c);
}

// ---------------------------------------------------------------------------
extern "C" void kernel_launch(void* const* d_in, const int* in_sizes, int n_in,
                              void* d_out, int out_size, void* d_ws, size_t ws_size,
                              hipStream_t stream) {
    const float* x   = (const float*)d_in[0];
    const int*   ei  = (const int*)  d_in[1];
    const float* ew  = (const float*)d_in[2];
    const float* W1  = (const float*)d_in[3];
    const float* a1s = (const float*)d_in[4];
    const float* a1d = (const float*)d_in[5];
    const float* W2  = (const float*)d_in[6];
    const float* a2s = (const float*)d_in[7];
    const float* a2d = (const float*)d_in[8];

    const int HID = in_sizes[4];              // 64
    const int FIN = in_sizes[3] / HID;        // 256
    const int NC  = in_sizes[7];              // 40
    const int N   = in_sizes[0] / FIN;        // 100000
    const int E   = in_sizes[2];              // 1600000
    const int EA  = E + N;                    // edges + self loops

    float*    h1    = (float*)d_ws;                    // N*HID
    float*    out1  = h1   + (size_t)N * HID;          // N*HID
    float*    h2    = out1 + (size_t)N * HID;          // N*NC
    float*    sArr  = h2   + (size_t)N * NC;           // N
    float*    dArr  = sArr + N;                        // N
    float*    denom = dArr + N;                        // N
    unsigned* mArr  = (unsigned*)(denom + N);          // N
    float*    exw   = (float*)(mArr + N);              // EA
    float*    outF  = (float*)d_out;                   // N*NC

    const int nrt     = (N + 15) / 16;
    const int pairs   = (nrt + 1) / 2;
    const int gBlocks = (pairs + 7) / 8;
    const int eBlocks = (EA + 255) / 256;
    const int nBlocks = (N + 255) / 256;

    // ---------------- layer 1 ----------------
    gemm1_kernel<<<gBlocks, 256, 0, stream>>>(x, W1, h1, N, FIN);
    sd_kernel<<<((long long)N * 32 + 255) / 256, 256, 0, stream>>>(h1, a1s, a1d, sArr, dArr, N, HID);
    init_seg<<<nBlocks, 256, 0, stream>>>(mArr, denom, N);
    zero_f32<<<(int)(((long long)N * HID + 255) / 256), 256, 0, stream>>>(out1, (long long)N * HID);
    edge_max_kernel<<<eBlocks, 256, 0, stream>>>(ei, sArr, dArr, mArr, E, EA);
    edge_expsum_kernel<<<eBlocks, 256, 0, stream>>>(ei, ew, sArr, dArr, mArr, denom, exw, E, EA);
    edge_agg_kernel<<<(int)(((long long)EA * 64 + 255) / 256), 256, 0, stream>>>(
        ei, exw, denom, h1, out1, E, EA, HID);

    // ---------------- layer 2 ---------------- (ReLU fused into GEMM2 A-load)
    gemm2_kernel<<<gBlocks, 256, 0, stream>>>(out1, W2, h2, N, NC);
    sd_kernel<<<((long long)N * 32 + 255) / 256, 256, 0, stream>>>(h2, a2s, a2d, sArr, dArr, N, NC);
    init_seg<<<nBlocks, 256, 0, stream>>>(mArr, denom, N);
    zero_f32<<<(int)(((long long)N * NC + 255) / 256), 256, 0, stream>>>(outF, (long long)N * NC);
    edge_max_kernel<<<eBlocks, 256, 0, stream>>>(ei, sArr, dArr, mArr, E, EA);
    edge_expsum_kernel<<<eBlocks, 256, 0, stream>>>(ei, ew, sArr, dArr, mArr, denom, exw, E, EA);
    edge_agg_kernel<<<(int)(((long long)EA * 64 + 255) / 256), 256, 0, stream>>>(
        ei, exw, denom, h2, outF, E, EA, NC);
}